// Lipschitz_loss_43542378447385
// MI455X (gfx1250) — compile-verified
//
#include <hip/hip_runtime.h>
#include <hip/hip_bf16.h>

typedef __attribute__((ext_vector_type(2))) float v2f;
typedef __attribute__((ext_vector_type(8))) float v8f;

#define EPSV       1e-10f
#define COS_EPSV   1e-8f
#define NBATCH     256
#define NFRAME     4
#define FEATD      512
#define IMG_ELEMS  37632      // 3*112*112
#define IMG_V4     9408       // IMG_ELEMS/4
#define CHUNKS     4
#define CHUNK_V4   2352       // IMG_V4/CHUNKS
#define K1_THREADS 256

// ---------------------------------------------------------------------------
// Kernel 1: streaming reduction over img. One block per (batch, chunk).
// Reads all 4 frames once per element position (float4 / global_load_b128),
// accumulates 3 squared-diff sums + 3 "any != frame0" flags.
// ---------------------------------------------------------------------------
__global__ void __launch_bounds__(K1_THREADS)
img_reduce_kernel(const float* __restrict__ img,
                  float* __restrict__ part_sum,     // [NBATCH][CHUNKS][3]
                  unsigned* __restrict__ part_neq)  // [NBATCH][CHUNKS][3]
{
    const int b   = blockIdx.x;
    const int c   = blockIdx.y;
    const int tid = threadIdx.x;

    const float4* f0 = (const float4*)(img + (size_t)(b * NFRAME + 0) * IMG_ELEMS);
    const float4* f1 = (const float4*)(img + (size_t)(b * NFRAME + 1) * IMG_ELEMS);
    const float4* f2 = (const float4*)(img + (size_t)(b * NFRAME + 2) * IMG_ELEMS);
    const float4* f3 = (const float4*)(img + (size_t)(b * NFRAME + 3) * IMG_ELEMS);

    float    s0 = 0.f, s1 = 0.f, s2 = 0.f;
    unsigned n1 = 0u,  n2 = 0u,  n3 = 0u;

    const int qend = (c + 1) * CHUNK_V4;
    for (int q = c * CHUNK_V4 + tid; q < qend; q += K1_THREADS) {
        float4 v0 = f0[q];
        float4 v1 = f1[q];
        float4 v2 = f2[q];
        float4 v3 = f3[q];
        float d;
        d = v1.x - v0.x; s0 = fmaf(d, d, s0);
        d = v1.y - v0.y; s0 = fmaf(d, d, s0);
        d = v1.z - v0.z; s0 = fmaf(d, d, s0);
        d = v1.w - v0.w; s0 = fmaf(d, d, s0);
        d = v2.x - v1.x; s1 = fmaf(d, d, s1);
        d = v2.y - v1.y; s1 = fmaf(d, d, s1);
        d = v2.z - v1.z; s1 = fmaf(d, d, s1);
        d = v2.w - v1.w; s1 = fmaf(d, d, s1);
        d = v3.x - v2.x; s2 = fmaf(d, d, s2);
        d = v3.y - v2.y; s2 = fmaf(d, d, s2);
        d = v3.z - v2.z; s2 = fmaf(d, d, s2);
        d = v3.w - v2.w; s2 = fmaf(d, d, s2);
        n1 |= (unsigned)((v1.x != v0.x) | (v1.y != v0.y) | (v1.z != v0.z) | (v1.w != v0.w));
        n2 |= (unsigned)((v2.x != v0.x) | (v2.y != v0.y) | (v2.z != v0.z) | (v2.w != v0.w));
        n3 |= (unsigned)((v3.x != v0.x) | (v3.y != v0.y) | (v3.z != v0.z) | (v3.w != v0.w));
    }

    __shared__ float    ss[3][K1_THREADS];
    __shared__ unsigned sn[3][K1_THREADS];
    ss[0][tid] = s0; ss[1][tid] = s1; ss[2][tid] = s2;
    sn[0][tid] = n1; sn[1][tid] = n2; sn[2][tid] = n3;
    __syncthreads();
    for (int st = K1_THREADS / 2; st > 0; st >>= 1) {
        if (tid < st) {
            ss[0][tid] += ss[0][tid + st];
            ss[1][tid] += ss[1][tid + st];
            ss[2][tid] += ss[2][tid + st];
            sn[0][tid] |= sn[0][tid + st];
            sn[1][tid] |= sn[1][tid + st];
            sn[2][tid] |= sn[2][tid + st];
        }
        __syncthreads();
    }
    if (tid == 0) {
        const int base = (b * CHUNKS + c) * 3;
        part_sum[base + 0] = ss[0][0];
        part_sum[base + 1] = ss[1][0];
        part_sum[base + 2] = ss[2][0];
        part_neq[base + 0] = sn[0][0];
        part_neq[base + 1] = sn[1][0];
        part_neq[base + 2] = sn[2][0];
    }
}

// ---------------------------------------------------------------------------
// Kernel 2: per 4-batch group (one wave32), compute the 16x16 Gram matrix of
// the group's 16x512 feature block via V_WMMA_F32_16X16X4_F32 (exact f32),
// extract diag (norms^2) + superdiagonal (dots), combine with img partials,
// produce per-batch penalty and the adaptive-weight output vector.
//
// A-fragment layout (32-bit A 16x4): lane<16 -> M=lane, K={k0,k0+1};
// lane>=16 -> M=lane-16, K={k0+2,k0+3}. For B = A^T the 4x16 B fragment has
// the identical per-lane contents, so one fragment feeds both operands.
// ---------------------------------------------------------------------------
__global__ void __launch_bounds__(32)
feat_gram_kernel(const float* __restrict__ feat,       // [NBATCH][4][512]
                 const float* __restrict__ feat_norm,  // [NBATCH][4][1]
                 const float* __restrict__ part_sum,
                 const unsigned* __restrict__ part_neq,
                 float* __restrict__ penalty,          // [NBATCH]
                 float* __restrict__ out_w)            // d_out+1, [NBATCH]
{
    const int g    = blockIdx.x;       // group of 4 batches
    const int lane = threadIdx.x;
    const int m    = lane & 15;        // row within group (0..15)
    const int hi   = (lane >= 16) ? 1 : 0;

    const float* rowp = feat + ((size_t)(g * 16 + m)) * FEATD + 2 * hi;

    v8f acc = {};
#pragma unroll 4
    for (int k0 = 0; k0 < FEATD; k0 += 4) {
        v2f a = *(const v2f*)(rowp + k0);
        acc = __builtin_amdgcn_wmma_f32_16x16x4_f32(
            /*neg_a=*/false, a, /*neg_b=*/false, a,
            /*c_mod=*/(short)0, acc, /*reuse_a=*/false, /*reuse_b=*/false);
    }

    // C/D layout: VGPR r, lanes 0-15 -> M=r, lanes 16-31 -> M=r+8; N = lane&15.
    __shared__ float gdiag[16];
    __shared__ float goff[16];
#pragma unroll
    for (int r = 0; r < 8; ++r) {
        const int M = r + (hi ? 8 : 0);
        if (m == M) gdiag[M] = acc[r];
        if (m == M + 1 && (M & 3) != 3) goff[M] = acc[r];
    }
    __syncthreads();

    if (lane < 4) {
        const int b  = g * 4 + lane;
        const float fn0  = feat_norm[(size_t)b * NFRAME];   // [b,0,0]
        const bool  cond = fn0 > 0.0f;
        const float w    = 1.0f / (__expf(fn0) + EPSV);

        float pen = 0.0f;
#pragma unroll
        for (int j = 0; j < 3; ++j) {
            const int M = 4 * lane + j;
            const float na  = fmaxf(sqrtf(gdiag[M]),     COS_EPSV);
            const float nb  = fmaxf(sqrtf(gdiag[M + 1]), COS_EPSV);
            const float cosv = goff[M] / (na * nb);

            const int base = b * CHUNKS * 3 + j;
            float sumsq = part_sum[base + 0 * 3] + part_sum[base + 1 * 3] +
                          part_sum[base + 2 * 3] + part_sum[base + 3 * 3];
            unsigned neq = part_neq[base + 0 * 3] | part_neq[base + 1 * 3] |
                           part_neq[base + 2 * 3] | part_neq[base + 3 * 3];

            const float idiff = sqrtf(sumsq) + EPSV;
            const float ratio = (1.0f - cosv) / idiff;   // LIP = 0
            const float term  = fmaxf(ratio, 0.0f) * w;  // SQUARED = false
            if (cond && neq) pen += term;
        }
        penalty[b] = pen;
        out_w[b]   = cond ? w : 0.0f;
    }
}

// ---------------------------------------------------------------------------
// Kernel 3: mean of penalty over 256 batches -> d_out[0]. LAMB_LIP = 1.
// ---------------------------------------------------------------------------
__global__ void __launch_bounds__(256)
mean_kernel(const float* __restrict__ penalty, float* __restrict__ out0)
{
    const int tid = threadIdx.x;
    __shared__ float s[256];
    s[tid] = penalty[tid];
    __syncthreads();
    for (int st = 128; st > 0; st >>= 1) {
        if (tid < st) s[tid] += s[tid + st];
        __syncthreads();
    }
    if (tid == 0) out0[0] = s[0] * (1.0f / (float)NBATCH);
}

extern "C" void kernel_launch(void* const* d_in, const int* in_sizes, int n_in,
                              void* d_out, int out_size, void* d_ws, size_t ws_size,
                              hipStream_t stream) {
    const float* img       = (const float*)d_in[0];  // (256,4,3,112,112)
    const float* feat      = (const float*)d_in[1];  // (256,4,512)
    const float* feat_norm = (const float*)d_in[2];  // (256,4,1)
    float* out = (float*)d_out;                      // [0]=loss, [1..256]=weights

    // Workspace layout (floats/uints):
    //   part_sum : NBATCH*CHUNKS*3 floats   @ 0
    //   part_neq : NBATCH*CHUNKS*3 uints    @ 3072
    //   penalty  : NBATCH floats            @ 6144
    float*    part_sum = (float*)d_ws;
    unsigned* part_neq = (unsigned*)d_ws + NBATCH * CHUNKS * 3;
    float*    penalty  = (float*)d_ws + 2 * NBATCH * CHUNKS * 3;

    dim3 g1(NBATCH, CHUNKS);
    img_reduce_kernel<<<g1, K1_THREADS, 0, stream>>>(img, part_sum, part_neq);

    feat_gram_kernel<<<NBATCH / 4, 32, 0, stream>>>(feat, feat_norm, part_sum,
                                                    part_neq, penalty, out + 1);

    mean_kernel<<<1, 256, 0, stream>>>(penalty, out);
}